// ENMessageBlock_34376918237206
// MI455X (gfx1250) — compile-verified
//
#include <hip/hip_runtime.h>
#include <math.h>

typedef __attribute__((ext_vector_type(2))) float v2f;
typedef __attribute__((ext_vector_type(8))) float v8f;
typedef int v4i_vs __attribute__((vector_size(16)));   // matches builtin param

#define FDIM 128
#define FDIM2 256
#define LDS_STRIDE 132   // 128 + 4 pad: A-tile rows land on distinct banks
#define ROWS 48          // rows per block = 3 x 16 row-tiles (amortizes B loads 3x)
#define RT 3

#define AS1 __attribute__((address_space(1)))
#define AS3 __attribute__((address_space(3)))

#if __has_builtin(__builtin_amdgcn_global_load_async_to_lds_b128) && \
    __has_builtin(__builtin_amdgcn_s_wait_asynccnt)
#define HAVE_ASYNC_LDS 1
#else
#define HAVE_ASYNC_LDS 0
#endif

// 16-byte global -> LDS copy; async (ASYNCcnt-tracked, no VGPR round trip) when
// the toolchain exposes the gfx1250 builtin, plain copy otherwise.
__device__ __forceinline__ void copy16_g2l(const float* __restrict__ g,
                                           float* __restrict__ l) {
#if HAVE_ASYNC_LDS
  __builtin_amdgcn_global_load_async_to_lds_b128(
      (AS1 v4i_vs*)g, (AS3 v4i_vs*)l, /*offset=*/0, /*cpol=*/0);
#else
  l[0] = g[0]; l[1] = g[1]; l[2] = g[2]; l[3] = g[3];
#endif
}

__device__ __forceinline__ void wait_g2l() {
#if HAVE_ASYNC_LDS
  __builtin_amdgcn_s_wait_asynccnt(0);
#endif
}

__device__ __forceinline__ v8f wmma_f32_k4(v2f a, v2f b, v8f c) {
  // D = A(16x4 f32) x B(4x16 f32) + C(16x16 f32); exact fp32 on the matrix pipe
  return __builtin_amdgcn_wmma_f32_16x16x4_f32(
      false, a, false, b, (short)0, c, false, false);
}

// ---------------------------------------------------------------------------
// Kernel 1: phi_all[n, 0:256] = silu(s @ W1 + b1) @ W2 + b2
// 48 rows per block (3 row-tiles), 8 waves. Each wave owns one 16-col tile of
// h (stage A) and col-tiles {w, w+128} of phi (stage B); each B fetch feeds
// three WMMAs (one per row-tile).
// ---------------------------------------------------------------------------
__global__ __launch_bounds__(256) void node_gemm_kernel(
    const float* __restrict__ s, const float* __restrict__ W1,
    const float* __restrict__ b1, const float* __restrict__ W2,
    const float* __restrict__ b2, float* __restrict__ phi_all, int nrows)
{
  __shared__ float s_lds[ROWS * LDS_STRIDE];
  __shared__ float h_lds[ROWS * LDS_STRIDE];

  const int tid  = threadIdx.x;
  const int wave = tid >> 5;
  const int lane = tid & 31;
  const int row0 = blockIdx.x * ROWS;

  // ---- Stage 0: A tile (48x128) global -> LDS ----
  if (row0 + ROWS <= nrows) {
    // full tile: async 16B chunks (32 chunks per 128-float row)
    for (int c = tid; c < ROWS * FDIM / 4; c += 256) {
      const int r = c >> 5;
      const int col = (c & 31) * 4;
      copy16_g2l(s + (size_t)(row0 + r) * FDIM + col,
                 &s_lds[r * LDS_STRIDE + col]);
    }
  } else {
    for (int i = tid; i < ROWS * FDIM; i += 256) {
      const int r = i >> 7, c = i & 127;
      const int gr = row0 + r;
      s_lds[r * LDS_STRIDE + c] = (gr < nrows) ? s[(size_t)gr * FDIM + c] : 0.0f;
    }
  }
  wait_g2l();
  __syncthreads();

  // WMMA lane mapping (ISA 7.12.2, 32-bit A 16x4 / B 4x16 / C 16x16):
  //  A: lanes 0-15 -> M=lane {K=kb,kb+1}; lanes 16-31 -> M=lane-16 {kb+2,kb+3}
  //  B: VGPR0: K=kb / kb+2; VGPR1: K=kb+1 / kb+3; N = lane&15
  //  C: VGPR r: M = r + (lane>>4)*8, N = lane&15
  const int lrow  = lane & 15;
  const int khalf = (lane >> 4) * 2;
  const int rbase = (lane >> 4) * 8;

  // ---- Stage A: h = silu(s @ W1 + b1); wave owns cols [16w, 16w+16) ----
  {
    const int col = 16 * wave + lrow;
    v8f acc[RT] = {};
    for (int kb = 0; kb < FDIM; kb += 4) {
      v2f b;
      b.x = W1[(size_t)(kb + khalf) * FDIM + col];
      b.y = W1[(size_t)(kb + khalf + 1) * FDIM + col];
#pragma unroll
      for (int t = 0; t < RT; ++t) {
        v2f a;
        a.x = s_lds[(t * 16 + lrow) * LDS_STRIDE + kb + khalf];
        a.y = s_lds[(t * 16 + lrow) * LDS_STRIDE + kb + khalf + 1];
        acc[t] = wmma_f32_k4(a, b, acc[t]);
      }
    }
    const float bias = b1[col];
#pragma unroll
    for (int t = 0; t < RT; ++t)
#pragma unroll
      for (int r = 0; r < 8; ++r) {
        const float x = acc[t][r] + bias;
        const float sig = 1.0f / (1.0f + __expf(-x));
        h_lds[(t * 16 + rbase + r) * LDS_STRIDE + col] = x * sig;
      }
  }
  __syncthreads();

  // ---- Stage B: phi = h @ W2 + b2; wave owns cols {16w..16w+15, +128} ----
  {
    const int colA = 16 * wave + lrow;
    const int colB = colA + 128;
    v8f acc[RT][2] = {};
    for (int kb = 0; kb < FDIM; kb += 4) {
      v2f bA, bB;
      bA.x = W2[(size_t)(kb + khalf) * FDIM2 + colA];
      bA.y = W2[(size_t)(kb + khalf + 1) * FDIM2 + colA];
      bB.x = W2[(size_t)(kb + khalf) * FDIM2 + colB];
      bB.y = W2[(size_t)(kb + khalf + 1) * FDIM2 + colB];
#pragma unroll
      for (int t = 0; t < RT; ++t) {
        v2f a;
        a.x = h_lds[(t * 16 + lrow) * LDS_STRIDE + kb + khalf];
        a.y = h_lds[(t * 16 + lrow) * LDS_STRIDE + kb + khalf + 1];
        acc[t][0] = wmma_f32_k4(a, bA, acc[t][0]);
        acc[t][1] = wmma_f32_k4(a, bB, acc[t][1]);
      }
    }
    const float biasA = b2[colA];
    const float biasB = b2[colB];
#pragma unroll
    for (int t = 0; t < RT; ++t)
#pragma unroll
      for (int r = 0; r < 8; ++r) {
        const int grow = row0 + t * 16 + rbase + r;
        if (grow < nrows) {
          phi_all[(size_t)grow * FDIM2 + colA] = acc[t][0][r] + biasA;
          phi_all[(size_t)grow * FDIM2 + colB] = acc[t][1][r] + biasB;
        }
      }
  }
}

// ---------------------------------------------------------------------------
// Kernel 2: per-edge RBF filter, gather phi[src], scatter-add into outputs.
// Wd/bd staged in LDS once per block; each wave processes 8 edges,
// 8 waves -> 64 edges per block.
// ---------------------------------------------------------------------------
#define EPW 8   // edges per wave

__global__ __launch_bounds__(256) void edge_kernel(
    const float* __restrict__ r_ij, const int* __restrict__ nbrs,
    const float* __restrict__ Wd, const float* __restrict__ bd,
    const float* __restrict__ phi_all,
    float* __restrict__ delta_s, float* __restrict__ delta_v, int nedges)
{
  __shared__ float wd_lds[20 * FDIM2];  // 20 KB
  __shared__ float bd_lds[FDIM2];       // 1 KB

  const int tid  = threadIdx.x;
  const int wave = tid >> 5;
  const int lane = tid & 31;

  // Stage filter weights: 20*256 + 256 floats, 16B chunks.
  for (int c = tid; c < 20 * FDIM2 / 4; c += 256)
    copy16_g2l(Wd + c * 4, &wd_lds[c * 4]);
  if (tid < FDIM2 / 4)
    copy16_g2l(bd + tid * 4, &bd_lds[tid * 4]);
  wait_g2l();
  __syncthreads();

  const int e0 = (blockIdx.x * 8 + wave) * EPW;

  for (int i = 0; i < EPW; ++i) {
    const int e = e0 + i;
    if (e >= nedges) break;

    const int dest = nbrs[2 * e + 0];
    const int src  = nbrs[2 * e + 1];

    const float r0 = r_ij[3 * (size_t)e + 0];
    const float r1 = r_ij[3 * (size_t)e + 1];
    const float r2 = r_ij[3 * (size_t)e + 2];
    const float dist  = sqrtf(r0 * r0 + r1 * r1 + r2 * r2 + 3e-8f);
    const float inv_d = 1.0f / dist;
    const float u0 = r0 * inv_d, u1 = r1 * inv_d, u2 = r2 * inv_d;

    const float t   = 3.14159265358979323f * dist * 0.2f;  // pi*dist/CUTOFF
    const float env = (dist < 5.0f) ? 0.5f * (__cosf(t) + 1.0f) : 0.0f;

    float rbf[20];
#pragma unroll
    for (int k = 0; k < 20; ++k)
      rbf[k] = __sinf((float)(k + 1) * t) * inv_d;

    const float* __restrict__ phi_row = phi_all + (size_t)src * FDIM2;
    float* __restrict__ dvrow = delta_v + (size_t)dest * FDIM * 3;
    float* __restrict__ dsrow = delta_s + (size_t)dest * FDIM;

#pragma unroll
    for (int j = 0; j < 8; ++j) {
      const int col = lane + 32 * j;  // coalesced; branch below is wave-uniform
      float w = bd_lds[col];
#pragma unroll
      for (int k = 0; k < 20; ++k)
        w = fmaf(rbf[k], wd_lds[k * FDIM2 + col], w);
      w *= env;
      const float inv = phi_row[col] * w;
      if (col < FDIM) {
        float* dv = dvrow + 3 * col;   // equi filter -> delta_v[dest, col, 0:3]
        unsafeAtomicAdd(dv + 0, inv * u0);
        unsafeAtomicAdd(dv + 1, inv * u1);
        unsafeAtomicAdd(dv + 2, inv * u2);
      } else {
        unsafeAtomicAdd(dsrow + (col - FDIM), inv);  // -> delta_s[dest, col-128]
      }
    }
  }
}

extern "C" void kernel_launch(void* const* d_in, const int* in_sizes, int n_in,
                              void* d_out, int out_size, void* d_ws, size_t ws_size,
                              hipStream_t stream) {
  const float* s_j  = (const float*)d_in[0];
  // d_in[1] = v_j : unused by the reference computation
  const float* r_ij = (const float*)d_in[2];
  const int*   nbrs = (const int*)d_in[3];
  const float* W1 = (const float*)d_in[4];
  const float* b1 = (const float*)d_in[5];
  const float* W2 = (const float*)d_in[6];
  const float* b2 = (const float*)d_in[7];
  const float* Wd = (const float*)d_in[8];
  const float* bd = (const float*)d_in[9];

  const int N = in_sizes[0] / FDIM;  // 50000
  const int E = in_sizes[2] / 3;     // 800000

  float* out     = (float*)d_out;
  float* delta_s = out;                     // N*128
  float* delta_v = out + (size_t)N * FDIM;  // N*128*3

  float* phi_all = (float*)d_ws;            // N*256 floats of scratch

  // Outputs are accumulated via atomics -> zero them every call (capturable).
  (void)hipMemsetAsync(d_out, 0, (size_t)out_size * sizeof(float), stream);

  node_gemm_kernel<<<(N + ROWS - 1) / ROWS, 256, 0, stream>>>(
      s_j, W1, b1, W2, b2, phi_all, N);
  edge_kernel<<<(E + 8 * EPW - 1) / (8 * EPW), 256, 0, stream>>>(
      r_ij, nbrs, Wd, bd, phi_all, delta_s, delta_v, E);
}